// ConvexDecoder_11141145166109
// MI455X (gfx1250) — compile-verified
//
#include <hip/hip_runtime.h>
#include <math.h>

#ifndef M_PI
#define M_PI 3.14159265358979323846
#endif

#define BQ 8
#define PQ 16
#define VQ 32
#define DQ 686
#define NPCQ 16384
#define NNSQ 8192
#define NOUTQ 8192
#define BP (BQ*PQ)
#define PD (PQ*DQ)

typedef float v2f __attribute__((ext_vector_type(2)));
typedef float v8f __attribute__((ext_vector_type(8)));

__device__ __forceinline__ float clampf(float x, float lo, float hi) {
  return fminf(fmaxf(x, lo), hi);
}
// x > 0 guaranteed by callers (clipped to >= 1e-30)
__device__ __forceinline__ float powp(float x, float p) {
  return exp2f(p * log2f(x));
}

// ---------------------------------------------------------------------------
// DIRS generation (matches numpy f64 linspace/cos/sin, cast to f32)
// ---------------------------------------------------------------------------
__global__ void k_dirs(float* __restrict__ dirs) {
  int idx = blockIdx.x * blockDim.x + threadIdx.x;
  if (idx >= DQ) return;
  int i, j;
  if (idx < 18 * 38)      { i = idx / 38 + 1; j = idx % 38; }
  else if (idx == 684)    { i = 0;  j = 0; }
  else                    { i = 19; j = 0; }
  double step = M_PI / 19.0;                      // common step of th1 and th2
  double th1 = (i == 19) ? (M_PI * 0.5) : (-M_PI * 0.5 + (double)i * step);
  double th2 = -M_PI + (double)j * step;
  double c1 = cos(th1), s1 = sin(th1);
  double c2 = cos(th2), s2 = sin(th2);
  dirs[idx * 3 + 0] = (float)(c1 * c2);
  dirs[idx * 3 + 1] = (float)(c1 * s2);
  dirs[idx * 3 + 2] = (float)s1;
}

// ---------------------------------------------------------------------------
// Per-primitive mean + local vertices. One wave (32 lanes = 32 verts) per bp.
// ---------------------------------------------------------------------------
__global__ void k_prep(const float* __restrict__ verts,
                       float* __restrict__ lv, float* __restrict__ mean) {
  int bp = blockIdx.x;
  int v  = threadIdx.x;              // 0..31
  const float* s = verts + (bp * VQ + v) * 3;
  float x = s[0], y = s[1], z = s[2];
  float sx = x, sy = y, sz = z;
  for (int o = 16; o >= 1; o >>= 1) {
    sx += __shfl_xor(sx, o, 32);
    sy += __shfl_xor(sy, o, 32);
    sz += __shfl_xor(sz, o, 32);
  }
  float mx = sx * (1.0f / VQ), my = sy * (1.0f / VQ), mz = sz * (1.0f / VQ);
  float* d = lv + (bp * VQ + v) * 3;
  d[0] = x - mx; d[1] = y - my; d[2] = z - mz;
  if (v == 0) { mean[bp * 3 + 0] = mx; mean[bp * 3 + 1] = my; mean[bp * 3 + 2] = mz; }
}

// ---------------------------------------------------------------------------
// Scalar support value h_out = clip(h/k) for one direction (used by the small
// kernels: h over DIRS, overlap).  lv points at 32x3 floats (LDS).
// ---------------------------------------------------------------------------
__device__ float support_hk_scalar(const float* lv, float p, float pinv,
                                   float dx, float dy, float dz) {
  float z[VQ];
  float zmax = -3.0e38f;
#pragma unroll
  for (int v = 0; v < VQ; ++v) {
    z[v] = lv[v*3+0]*dx + lv[v*3+1]*dy + lv[v*3+2]*dz;
    zmax = fmaxf(zmax, z[v]);
  }
  float maxz = clampf(zmax, 1e-30f, 1e30f);
  float k = 1.0f / maxz;
  float s = 0.0f;
#pragma unroll
  for (int v = 0; v < VQ; ++v) {
    float zmk = clampf(fmaxf(z[v], 0.0f) * k, 1e-30f, 1e30f);
    s += powp(zmk, p);
  }
  float h = exp2f(log2f(s) * pinv);         // s==0 -> -inf -> 0, matches 0**(1/p)
  return fminf(h * maxz, 1e30f);            // h_out = clip(h/k)
}

// ---------------------------------------------------------------------------
// h0[bp,d] over DIRS + write x output (broadcast DIRS)
// ---------------------------------------------------------------------------
__global__ void k_support_dirs(const float* __restrict__ dirs,
                               const float* __restrict__ lv,
                               const float* __restrict__ smooth,
                               float* __restrict__ h0, float* __restrict__ x_out) {
  __shared__ float slv[VQ * 3];
  __shared__ float sp, spinv;
  int bp = blockIdx.y;
  int t  = threadIdx.x;
  if (t < VQ * 3) slv[t] = lv[bp * VQ * 3 + t];
  if (t == 0) { sp = smooth[bp]; spinv = 1.0f / smooth[bp]; }
  __syncthreads();
  int d = blockIdx.x * blockDim.x + t;
  if (d >= DQ) return;
  float dx = dirs[d*3+0], dy = dirs[d*3+1], dz = dirs[d*3+2];
  h0[bp * DQ + d] = support_hk_scalar(slv, sp, spinv, dx, dy, dz);
  float* xo = x_out + (bp * DQ + d) * 3;
  xo[0] = dx; xo[1] = dy; xo[2] = dz;
}

// ---------------------------------------------------------------------------
// scale[bp] = clip(max_d h0, 1e-10, 10)
// ---------------------------------------------------------------------------
__global__ void k_scale(const float* __restrict__ h0, float* __restrict__ scale) {
  __shared__ float red[256];
  int bp = blockIdx.x, t = threadIdx.x;
  float m = -3.0e38f;
  for (int d = t; d < DQ; d += 256) m = fmaxf(m, h0[bp * DQ + d]);
  red[t] = m; __syncthreads();
  for (int o = 128; o >= 1; o >>= 1) {
    if (t < o) red[t] = fmaxf(red[t], red[t + o]);
    __syncthreads();
  }
  if (t == 0) scale[bp] = clampf(red[0], 1e-10f, 10.0f);
}

// ---------------------------------------------------------------------------
// Surf points (dhdz einsum) + normals for the filter.
// ---------------------------------------------------------------------------
__global__ void k_surfpts(const float* __restrict__ dirs, const float* __restrict__ lv,
                          const float* __restrict__ mean, const float* __restrict__ smooth,
                          const float* __restrict__ scale,
                          float* __restrict__ normals, float* __restrict__ surf_out) {
  __shared__ float slv[VQ * 3];
  __shared__ float sm[3];
  __shared__ float sp, spinv, ssc;
  int bp = blockIdx.y;
  int t  = threadIdx.x;
  if (t < VQ * 3) slv[t] = lv[bp * VQ * 3 + t];
  if (t < 3) sm[t] = mean[bp * 3 + t];
  if (t == 0) { sp = smooth[bp]; spinv = 1.0f / smooth[bp]; ssc = scale[bp]; }
  __syncthreads();
  int d = blockIdx.x * blockDim.x + t;
  if (d >= DQ) return;
  float lx = dirs[d*3+0] * ssc, ly = dirs[d*3+1] * ssc, lz = dirs[d*3+2] * ssc;
  float nsq = clampf(lx*lx + ly*ly + lz*lz, 1e-40f, 1e40f);
  float rinv = 1.0f / sqrtf(nsq);
  float dx = lx * rinv, dy = ly * rinv, dz = lz * rinv;
  float* nv = normals + (bp * DQ + d) * 3;
  nv[0] = dx; nv[1] = dy; nv[2] = dz;

  float z[VQ]; float zmax = -3.0e38f;
#pragma unroll
  for (int v = 0; v < VQ; ++v) {
    z[v] = slv[v*3+0]*dx + slv[v*3+1]*dy + slv[v*3+2]*dz;
    zmax = fmaxf(zmax, z[v]);
  }
  float maxz = clampf(zmax, 1e-30f, 1e30f);
  float k = 1.0f / maxz;
  float zmk[VQ]; float s = 0.0f;
#pragma unroll
  for (int v = 0; v < VQ; ++v) {
    zmk[v] = clampf(fmaxf(z[v], 0.0f) * k, 1e-30f, 1e30f);
    s += powp(zmk[v], sp);
  }
  float h  = exp2f(log2f(s) * spinv);
  float hsinv = 1.0f / clampf(h, 1e-30f, 1e30f);
  float pm1 = sp - 1.0f;
  float ax = 0.f, ay = 0.f, az = 0.f;
#pragma unroll
  for (int v = 0; v < VQ; ++v) {
    float w = clampf(powp(zmk[v] * hsinv, pm1), 1e-30f, 1e30f);   // dhdz
    ax += w * slv[v*3+0]; ay += w * slv[v*3+1]; az += w * slv[v*3+2];
  }
  float* so = surf_out + (bp * DQ + d) * 3;   // (B, P*D, 3) flat
  so[0] = ax + sm[0]; so[1] = ay + sm[1]; so[2] = az + sm[2];
}

// ---------------------------------------------------------------------------
// Overlap: one block per batch b; thread t -> (i = t/16, j = t%16).
// ---------------------------------------------------------------------------
__global__ void k_overlap(const float* __restrict__ lv, const float* __restrict__ mean,
                          const float* __restrict__ smooth, float* __restrict__ ov) {
  __shared__ float slv[PQ * VQ * 3];   // 1536 f
  __shared__ float sm[PQ * 3];
  __shared__ float sp[PQ];
  __shared__ float hb[PQ * PQ];
  int b = blockIdx.x, t = threadIdx.x;
  for (int k = t; k < PQ * VQ * 3; k += 256) slv[k] = lv[b * PQ * VQ * 3 + k];
  if (t < PQ * 3) sm[t] = mean[b * PQ * 3 + t];
  if (t < PQ) sp[t] = smooth[b * PQ + t];
  __syncthreads();
  int i = t >> 4, j = t & 15;
  float dxm = sm[j*3+0] - sm[i*3+0];
  float dym = sm[j*3+1] - sm[i*3+1];
  float dzm = sm[j*3+2] - sm[i*3+2];
  float dsq = clampf(dxm*dxm + dym*dym + dzm*dzm, 1e-20f, 1e20f);
  float dn  = sqrtf(dsq);
  float ux, uy, uz;
  if (i == j) { ux = 1.f; uy = 0.f; uz = 0.f; }
  else { float rinv = 1.0f / dn; ux = dxm * rinv; uy = dym * rinv; uz = dzm * rinv; }
  hb[t] = support_hk_scalar(&slv[i * VQ * 3], sp[i], 1.0f / sp[i], ux, uy, uz);
  __syncthreads();
  float sep = dn - hb[i * 16 + j] - hb[j * 16 + i];
  ov[b * 256 + t] = (i == j) ? 0.0f : fmaxf(-sep, 0.0f);
}

// ---------------------------------------------------------------------------
// WMMA workhorse helpers: lane pair (l, l^16) holds the 32 z values of point
// n = l&15 of its WMMA column group; one shfl_xor(16) merges the halves.
// ---------------------------------------------------------------------------
__device__ __forceinline__ float reduce_hk(v8f c0, v8f c1, float p, float pinv) {
  float zmax = c0[0];
#pragma unroll
  for (int i = 1; i < 8; ++i) zmax = fmaxf(zmax, c0[i]);
#pragma unroll
  for (int i = 0; i < 8; ++i) zmax = fmaxf(zmax, c1[i]);
  zmax = fmaxf(zmax, __shfl_xor(zmax, 16, 32));
  float maxz = clampf(zmax, 1e-30f, 1e30f);
  float k = 1.0f / maxz;
  float s = 0.0f;
#pragma unroll
  for (int i = 0; i < 8; ++i) {
    float zmk = clampf(fmaxf(c0[i], 0.0f) * k, 1e-30f, 1e30f);
    s += powp(zmk, p);
  }
#pragma unroll
  for (int i = 0; i < 8; ++i) {
    float zmk = clampf(fmaxf(c1[i], 0.0f) * k, 1e-30f, 1e30f);
    s += powp(zmk, p);
  }
  s += __shfl_xor(s, 16, 32);
  float h = exp2f(log2f(s) * pinv);
  return fminf(h * maxz, 1e30f);
}

// Main distance kernel: block = 8 waves x 32 points, one (b,p) per (z,y) block.
__global__ void k_dist(const float* __restrict__ pts, int npts, int apply_filter,
                       const float* __restrict__ lv, const float* __restrict__ mean,
                       const float* __restrict__ smooth, const float* __restrict__ normals,
                       float* __restrict__ outp) {
  __shared__ float slv[VQ * 3];
  __shared__ float sm[3];
  __shared__ float sp, spinv;
  int b = blockIdx.z, p = blockIdx.y;
  int bp = b * PQ + p;
  int t = threadIdx.x;
  if (t < VQ * 3) slv[t] = lv[bp * VQ * 3 + t];
  if (t < 3) sm[t] = mean[bp * 3 + t];
  if (t == 0) { sp = smooth[bp]; spinv = 1.0f / smooth[bp]; }
  __syncthreads();

  int lane = t & 31;
  int l16  = lane & 15;
  bool lo  = lane < 16;

  // A (16x4 f32): VGPR0 lanes0-15 K=0(x) / lanes16-31 K=2(z);
  //               VGPR1 lanes0-15 K=1(y) / lanes16-31 K=3(pad 0)
  v2f a0, a1;
  a0.x = lo ? slv[l16*3+0]      : slv[l16*3+2];
  a0.y = lo ? slv[l16*3+1]      : 0.0f;
  a1.x = lo ? slv[(l16+16)*3+0] : slv[(l16+16)*3+2];
  a1.y = lo ? slv[(l16+16)*3+1] : 0.0f;

  int pi = blockIdx.x * 256 + t;
  bool valid = pi < npts;
  int pidx = valid ? pi : 0;
  const float* pr = pts + (b * npts + pidx) * 3;
  float lx = pr[0] - sm[0], ly = pr[1] - sm[1], lz = pr[2] - sm[2];
  if (!valid) { lx = 1.0f; ly = 0.0f; lz = 0.0f; }   // safe dummy, store skipped
  float nsq = clampf(lx*lx + ly*ly + lz*lz, 1e-40f, 1e40f);
  float nrm = sqrtf(nsq);
  float rinv = 1.0f / nrm;                 // 1 division instead of 3
  float dx = lx * rinv, dy = ly * rinv, dz = lz * rinv;

  // B (4x16 f32, layout symmetric with A): build two 16-point column groups
  float s_dz   = __shfl(dz, l16, 32);
  float s_dx16 = __shfl(dx, l16 + 16, 32);
  float s_dy16 = __shfl(dy, l16 + 16, 32);
  v2f blo, bhi;
  blo.x = lo ? dx : s_dz;
  blo.y = lo ? dy : 0.0f;
  bhi.x = lo ? s_dx16 : dz;
  bhi.y = lo ? s_dy16 : 0.0f;

  v8f zero = {};
  v8f zlo0 = __builtin_amdgcn_wmma_f32_16x16x4_f32(false, a0, false, blo, (short)0, zero, false, false);
  v8f zlo1 = __builtin_amdgcn_wmma_f32_16x16x4_f32(false, a1, false, blo, (short)0, zero, false, false);
  v8f zhi0 = __builtin_amdgcn_wmma_f32_16x16x4_f32(false, a0, false, bhi, (short)0, zero, false, false);
  v8f zhi1 = __builtin_amdgcn_wmma_f32_16x16x4_f32(false, a1, false, bhi, (short)0, zero, false, false);

  float hk_lo = reduce_hk(zlo0, zlo1, sp, spinv);   // points 0-15 of this wave
  float hk_hi = reduce_hk(zhi0, zhi1, sp, spinv);   // points 16-31
  float hk = lo ? hk_lo : hk_hi;                    // each lane's own point
  float dist = nrm - hk;

  if (apply_filter && valid) {
    int j  = pi / DQ;
    int dd = pi - j * DQ;
    if (j == p) dist = 100.0f;               // diagonal override
    else {
      const float* mj = mean + (b * PQ + j) * 3;
      const float* nv = normals + (bp * DQ + dd) * 3;
      float nf = (mj[0]-sm[0])*nv[0] + (mj[1]-sm[1])*nv[1] + (mj[2]-sm[2])*nv[2];
      if (nf < 0.0f) dist = 100.0f;          // normal filter
    }
  }
  if (valid) outp[bp * npts + pi] = dist;
}

// ---------------------------------------------------------------------------
extern "C" void kernel_launch(void* const* d_in, const int* in_sizes, int n_in,
                              void* d_out, int out_size, void* d_ws, size_t ws_size,
                              hipStream_t stream) {
  (void)in_sizes; (void)n_in; (void)out_size; (void)ws_size;
  const float* verts  = (const float*)d_in[0];
  const float* smooth = (const float*)d_in[1];
  const float* pc     = (const float*)d_in[2];
  const float* nsf    = (const float*)d_in[3];
  const float* opts   = (const float*)d_in[4];
  float* out = (float*)d_out;
  float* ws  = (float*)d_ws;

  // workspace layout (floats)
  float* w_dirs  = ws;                    // D*3          = 2058
  float* w_lv    = w_dirs  + DQ * 3;      // BP*V*3       = 12288
  float* w_mean  = w_lv    + BP * VQ * 3; // BP*3         = 384
  float* w_h0    = w_mean  + BP * 3;      // BP*D         = 87808
  float* w_scale = w_h0    + BP * DQ;     // BP           = 128
  float* w_norm  = w_scale + BP;          // BP*D*3       = 263424

  // output layout (floats), tuple order
  float* o_overlap = out;                          // B*P*P      = 2048
  float* o_dist    = o_overlap + BQ * PQ * PQ;     // B*P*NPC    = 2097152
  float* o_x       = o_dist    + BP * NPCQ;        // B*P*D*3    = 263424
  float* o_sdist   = o_x       + BP * DQ * 3;      // B*P*(P*D)  = 1404928
  float* o_spts    = o_sdist   + BP * PD;          // B*(P*D)*3  = 263424
  float* o_nsd     = o_spts    + BQ * PD * 3;      // B*P*NNS    = 1048576
  float* o_od      = o_nsd     + BP * NNSQ;        // B*P*NOUT   = 1048576

  k_dirs<<<dim3((DQ + 255) / 256), 256, 0, stream>>>(w_dirs);
  k_prep<<<dim3(BP), 32, 0, stream>>>(verts, w_lv, w_mean);
  k_support_dirs<<<dim3((DQ + 255) / 256, BP), 256, 0, stream>>>(w_dirs, w_lv, smooth, w_h0, o_x);
  k_scale<<<dim3(BP), 256, 0, stream>>>(w_h0, w_scale);
  k_surfpts<<<dim3((DQ + 255) / 256, BP), 256, 0, stream>>>(w_dirs, w_lv, w_mean, smooth, w_scale, w_norm, o_spts);
  k_overlap<<<dim3(BQ), 256, 0, stream>>>(w_lv, w_mean, smooth, o_overlap);

  k_dist<<<dim3(NPCQ / 256, PQ, BQ), 256, 0, stream>>>(pc,    NPCQ, 0, w_lv, w_mean, smooth, w_norm, o_dist);
  k_dist<<<dim3((PD + 255) / 256, PQ, BQ), 256, 0, stream>>>(o_spts, PD, 1, w_lv, w_mean, smooth, w_norm, o_sdist);
  k_dist<<<dim3(NNSQ / 256, PQ, BQ), 256, 0, stream>>>(nsf,   NNSQ, 0, w_lv, w_mean, smooth, w_norm, o_nsd);
  k_dist<<<dim3(NOUTQ / 256, PQ, BQ), 256, 0, stream>>>(opts, NOUTQ, 0, w_lv, w_mean, smooth, w_norm, o_od);
}